// ZBLBasis_14035953123660
// MI455X (gfx1250) — compile-verified
//
#include <hip/hip_runtime.h>
#include <hip/hip_bf16.h>
#include <stdint.h>

#define N_ELEMS 10
#define TILE 256
#define WAVES_PER_BLOCK 8
#define BLOCK_THREADS (WAVES_PER_BLOCK * 32)

typedef uint32_t v4u __attribute__((ext_vector_type(4)));
typedef int      v4i __attribute__((ext_vector_type(4)));
typedef int      v8i __attribute__((ext_vector_type(8)));

#if defined(__HIP_DEVICE_COMPILE__) && __has_builtin(__builtin_amdgcn_tensor_load_to_lds) && __has_builtin(__builtin_amdgcn_s_wait_tensorcnt)
#define HAVE_TDM 1
#else
#define HAVE_TDM 0
#endif

// ---------------- Probe: surfaces the f32 atomic lowering at the top of the disasm ----
// (never launched; exists only so the snippet shows whether __hip_atomic_fetch_add
//  emits native global_atomic_add_f32 or a CAS loop)
__global__ void atomic_add_probe(float* __restrict__ out, const float* __restrict__ val) {
    (void)__hip_atomic_fetch_add(out + threadIdx.x, val[threadIdx.x],
                                 __ATOMIC_RELAXED, __HIP_MEMORY_SCOPE_AGENT);
}

// ---------------- TDM helper ----------------
#if HAVE_TDM
// 1-row tile descriptor copying `tile_elems` 4-byte elements from `gaddr` to LDS
// byte-offset `lds_off`. `valid_elems` (= tensor_dim0) clamps the tail: reads
// beyond tensor_dim0 return zero per ISA sec 8.2.
__device__ __forceinline__ void tdm_load_1d(uint32_t lds_off, uint64_t gaddr,
                                            uint32_t valid_elems, uint32_t tile_elems) {
    v4u g0;
    g0.x = 1u;                                                     // count=1, user mode
    g0.y = lds_off;                                                // lds_addr
    g0.z = (uint32_t)gaddr;                                        // global_addr[31:0]
    g0.w = ((uint32_t)(gaddr >> 32) & 0x01FFFFFFu) | (2u << 30);   // addr[56:32] | type=2
    v8i g1;
    g1[0] = (int)(2u << 16);                                       // data_size = 4B
    g1[1] = (int)((valid_elems & 0xFFFFu) << 16);                  // tensor_dim0 lo16
    g1[2] = (int)(((valid_elems >> 16) & 0xFFFFu) | (1u << 16));   // dim0 hi16 | tensor_dim1=1
    g1[3] = (int)((tile_elems & 0xFFFFu) << 16);                   // tile_dim0
    g1[4] = 1;                                                     // tile_dim1=1, tile_dim2=0
    g1[5] = (int)tile_elems;                                       // tensor_dim0_stride lo
    g1[6] = 0;
    g1[7] = 0;
    v4i gz4 = {0, 0, 0, 0};
    v8i gz8 = {0, 0, 0, 0, 0, 0, 0, 0};
    __builtin_amdgcn_tensor_load_to_lds(g0, g1, gz4, gz4, gz8, 0);
}
#endif

// ---------------- Kernel: edge phase ----------------
__global__ __launch_bounds__(BLOCK_THREADS)
void zbl_edge_kernel(const float* __restrict__ x,
                     const int* __restrict__ edge_index,
                     const float4* __restrict__ table,
                     float* __restrict__ out,
                     int n_edges) {
    const int* sender   = edge_index;
    const int* receiver = edge_index + n_edges;
    const float inv_apref = 1.0f / (0.4543f * 0.529f);

#if HAVE_TDM
    __shared__ float sx[2][WAVES_PER_BLOCK][TILE];
    __shared__ int   ssv[2][WAVES_PER_BLOCK][TILE];
    __shared__ int   srv[2][WAVES_PER_BLOCK][TILE];

    const int lane = (int)(threadIdx.x & 31u);
    const int wave = __builtin_amdgcn_readfirstlane((int)(threadIdx.x >> 5));
    const int gwave  = (int)blockIdx.x * WAVES_PER_BLOCK + wave;
    const int nwaves = (int)gridDim.x * WAVES_PER_BLOCK;
    const int ntiles = (n_edges + TILE - 1) / TILE;

    // LDS byte offsets of this wave's double buffers (flat addr low 32 bits == LDS offset)
    uint32_t xoff[2], soff[2], roff[2];
#pragma unroll
    for (int b = 0; b < 2; ++b) {
        xoff[b] = (uint32_t)(uintptr_t)&sx[b][wave][0];
        soff[b] = (uint32_t)(uintptr_t)&ssv[b][wave][0];
        roff[b] = (uint32_t)(uintptr_t)&srv[b][wave][0];
    }

    auto issue_tile = [&](int t, int b) {
        int tu = __builtin_amdgcn_readfirstlane(t);
        uint32_t base_e = (uint32_t)tu * TILE;
        uint32_t rem = (uint32_t)n_edges - base_e;
        uint32_t valid = rem < (uint32_t)TILE ? rem : (uint32_t)TILE;
        tdm_load_1d(xoff[b], (uint64_t)(uintptr_t)(x + base_e),        valid, TILE);
        tdm_load_1d(soff[b], (uint64_t)(uintptr_t)(sender + base_e),   valid, TILE);
        tdm_load_1d(roff[b], (uint64_t)(uintptr_t)(receiver + base_e), valid, TILE);
    };

    int buf = 0;
    if (gwave < ntiles) issue_tile(gwave, 0);
    for (int t = gwave; t < ntiles; t += nwaves) {
        int tn = t + nwaves;
        if (tn < ntiles) {
            issue_tile(tn, buf ^ 1);
            __builtin_amdgcn_s_wait_tensorcnt(3);   // current buf's 3 loads complete
        } else {
            __builtin_amdgcn_s_wait_tensorcnt(0);
        }
        int base = t * TILE;
#pragma unroll
        for (int k = lane; k < TILE; k += 32) {
            int e = base + k;
            if (e >= n_edges) break;
            float xe = sx[buf][wave][k];
            int s = ssv[buf][wave][k];
            int r = srv[buf][wave][k];
            float4 nu = table[s];
            float4 nv = table[r];
            float rmax = nu.z + nv.z;
            if (xe < rmax) {
                float roa = xe * (nu.y + nv.y) * inv_apref;   // x / a
                float phi = 0.1818f  * __expf(-3.2f    * roa)
                          + 0.5099f  * __expf(-0.9423f * roa)
                          + 0.2802f  * __expf(-0.4028f * roa)
                          + 0.02817f * __expf(-0.2016f * roa);
                float v = 14.3996f * nu.x * nv.x / xe * phi;
                float tq = xe / rmax;
                float t2 = tq * tq;
                float t6 = t2 * t2 * t2;
                float env = 1.0f + t6 * (-28.0f + tq * (48.0f - 21.0f * tq));
                (void)__hip_atomic_fetch_add(out + r, 0.5f * v * env,
                                             __ATOMIC_RELAXED, __HIP_MEMORY_SCOPE_AGENT);
            }
        }
        buf ^= 1;
    }
#else
    // Fallback: plain grid-stride streaming (still memory-optimal; no TDM builtin)
    int i = (int)(blockIdx.x * blockDim.x + threadIdx.x);
    int stride = (int)(gridDim.x * blockDim.x);
    for (int e = i; e < n_edges; e += stride) {
        float xe = x[e];
        int s = sender[e];
        int r = receiver[e];
        float4 nu = table[s];
        float4 nv = table[r];
        float rmax = nu.z + nv.z;
        if (xe < rmax) {
            float roa = xe * (nu.y + nv.y) * inv_apref;
            float phi = 0.1818f  * __expf(-3.2f    * roa)
                      + 0.5099f  * __expf(-0.9423f * roa)
                      + 0.2802f  * __expf(-0.4028f * roa)
                      + 0.02817f * __expf(-0.2016f * roa);
            float v = 14.3996f * nu.x * nv.x / xe * phi;
            float tq = xe / rmax;
            float t2 = tq * tq;
            float t6 = t2 * t2 * t2;
            float env = 1.0f + t6 * (-28.0f + tq * (48.0f - 21.0f * tq));
            (void)__hip_atomic_fetch_add(out + r, 0.5f * v * env,
                                         __ATOMIC_RELAXED, __HIP_MEMORY_SCOPE_AGENT);
        }
    }
#endif
}

// ---------------- Kernel: per-node precompute + output zeroing ----------------
// table[i] = { Z, Z^0.3, covalent_radii[Z], 0 }  (one random-gather record per node,
// 1.6 MB total -> stays resident in the 192MB L2 for the edge phase)
__global__ void node_precompute_kernel(const float* __restrict__ node_attrs,
                                       const int* __restrict__ atomic_numbers,
                                       const float* __restrict__ covalent_radii,
                                       float4* __restrict__ table,
                                       float* __restrict__ out,
                                       int n_nodes) {
    int i = blockIdx.x * blockDim.x + threadIdx.x;
    if (i >= n_nodes) return;
    const float* row = node_attrs + (size_t)i * N_ELEMS;
    float best = row[0];
    int arg = 0;
#pragma unroll
    for (int j = 1; j < N_ELEMS; ++j) {
        float v = row[j];
        if (v > best) { best = v; arg = j; }   // first-max, matches jnp.argmax
    }
    int Z = atomic_numbers[arg];
    float Zf = (float)Z;
    float Zp = powf(Zf, 0.3f);
    float rc = covalent_radii[Z];
    table[i] = make_float4(Zf, Zp, rc, 0.0f);
    out[i] = 0.0f;
}

extern "C" void kernel_launch(void* const* d_in, const int* in_sizes, int n_in,
                              void* d_out, int out_size, void* d_ws, size_t ws_size,
                              hipStream_t stream) {
    const float* x              = (const float*)d_in[0];
    const float* node_attrs     = (const float*)d_in[1];
    const int*   edge_index     = (const int*)d_in[2];
    const int*   atomic_numbers = (const int*)d_in[3];
    const float* covalent_radii = (const float*)d_in[4];
    float* out = (float*)d_out;
    float4* table = (float4*)d_ws;   // N_NODES * 16 bytes

    int n_edges = in_sizes[0];                 // x is [E,1]
    int n_nodes = in_sizes[1] / N_ELEMS;       // node_attrs is [N,10]

    node_precompute_kernel<<<(n_nodes + 255) / 256, 256, 0, stream>>>(
        node_attrs, atomic_numbers, covalent_radii, table, out, n_nodes);

    // 512 blocks x 8 waves -> ~3 tiles per wave: enough tiles for the TDM
    // double-buffer to overlap, and enough waves to fill the chip.
    zbl_edge_kernel<<<512, BLOCK_THREADS, 0, stream>>>(
        x, edge_index, table, out, n_edges);
}